// ATSSAssigner_46746424050471
// MI455X (gfx1250) — compile-verified
//
#include <hip/hip_runtime.h>
#include <hip/hip_bf16.h>
#include <math.h>

#define ATSS_TOPK 9
#define ATSS_EPS  1e-9f

typedef float v2f __attribute__((ext_vector_type(2)));
typedef float v8f __attribute__((ext_vector_type(8)));

// ---------------------------------------------------------------------------
// Kernel 0: init per-anchor scratch: claim count = 0, first claiming gt = +inf
// ---------------------------------------------------------------------------
__global__ __launch_bounds__(256) void atss_init_kernel(int* __restrict__ cnt,
                                                        int* __restrict__ firstgt,
                                                        int total) {
  int i = blockIdx.x * 256 + threadIdx.x;
  if (i < total) { cnt[i] = 0; firstgt[i] = 0x7fffffff; }
}

// ---------------------------------------------------------------------------
// Kernel 1: per 16-GT tile (one wave32), WMMA dist^2 tiles -> per-level top-9
// -> candidate IoU mean+std threshold -> scatter claims with atomics.
//   dist^2 = [gx, gy, |g|^2, 1] . [-2ax, -2ay, 1, |a|^2]  (K=4 f32 GEMM)
// ---------------------------------------------------------------------------
__global__ __launch_bounds__(32) void atss_topk_kernel(
    const float* __restrict__ anchors,   // [L,4]
    const float* __restrict__ gtb,       // [B,n,4]
    const float* __restrict__ padm,      // [B,n]
    int* __restrict__ cnt,               // [B,L]
    int* __restrict__ firstgt,           // [B,L]
    int B, int n, int L)
{
  constexpr int LBASE[3]  = {0, 6400, 8000};
  constexpr int LCOUNT[3] = {6400, 1600, 400};

  const int rowTiles = n >> 4;
  const int b    = blockIdx.x / rowTiles;
  const int rt   = blockIdx.x % rowTiles;
  const int lane = threadIdx.x;
  const int m    = lane & 15;   // column / row-in-halfwave index
  const int half = lane >> 4;   // 0: K=0,1 rows of A ; 1: K=2,3 rows
  const int gi   = rt * 16 + m; // gt row this lane represents (lanes 0-15)

  // gt box for row (lane&15)
  const float* g = gtb + ((size_t)(b * n + gi)) * 4;
  const float gx1 = g[0], gy1 = g[1], gx2 = g[2], gy2 = g[3];
  const float gcx = 0.5f * (gx1 + gx2), gcy = 0.5f * (gy1 + gy2);

  // A fragment (16x4 f32): lanes 0-15 -> K0,K1 ; lanes 16-31 -> K2,K3
  v2f af;
  af.x = half ? (gcx * gcx + gcy * gcy) : gcx;  // K0 = gx   | K2 = |g|^2
  af.y = half ? 1.0f                     : gcy; // K1 = gy   | K3 = 1

  __shared__ float tile[16][17];  // dist^2 tile, padded

  int cidx[27];  // candidate anchor indices (lanes 0-15 meaningful)

  #pragma unroll
  for (int lvl = 0; lvl < 3; ++lvl) {
    float tv[9]; int tix[9];
    #pragma unroll
    for (int k = 0; k < 9; ++k) { tv[k] = 3.0e38f; tix[k] = LBASE[lvl]; }

    const int base = LBASE[lvl], nl = LCOUNT[lvl];
    for (int tb = base; tb < base + nl; tb += 16) {
      // B fragment (4x16 f32): lanes 0-15 -> K0,K1 rows ; lanes 16-31 -> K2,K3
      const float* ab = anchors + (size_t)(tb + m) * 4;
      __builtin_prefetch(anchors + (size_t)(tb + 16 + m) * 4, 0, 3);
      const float ax1 = ab[0], ay1 = ab[1], ax2 = ab[2], ay2 = ab[3];
      const float acx = 0.5f * (ax1 + ax2), acy = 0.5f * (ay1 + ay2);
      v2f bf;
      bf.x = half ? 1.0f                     : (-2.0f * acx); // K0=-2ax | K2=1
      bf.y = half ? (acx * acx + acy * acy)  : (-2.0f * acy); // K1=-2ay | K3=|a|^2

      v8f cz = {};
      v8f d = __builtin_amdgcn_wmma_f32_16x16x4_f32(
          /*neg_a=*/false, af, /*neg_b=*/false, bf,
          /*c_mod=*/(short)0, cz, /*reuse_a=*/false, /*reuse_b=*/false);

      // C/D layout: lane<16: (M=v, N=lane) ; lane>=16: (M=v+8, N=lane-16)
      #pragma unroll
      for (int v = 0; v < 8; ++v) tile[v + 8 * half][m] = d[v];
      __syncthreads();

      if (half == 0) {
        // lane m owns gt row m: BRANCHLESS sorted-insert of 16 dist^2 values
        // into ascending top-9 (strict < keeps lax.top_k stable tie-break).
        #pragma unroll
        for (int j = 0; j < 16; ++j) {
          float dv = tile[m][j];
          int   di = tb + j;
          #pragma unroll
          for (int k = 0; k < 9; ++k) {
            const bool sw   = dv < tv[k];          // v_cmp
            const float lo  = fminf(dv, tv[k]);    // v_min_f32
            const float hi  = fmaxf(dv, tv[k]);    // v_max_f32
            const int   ilo = sw ? di : tix[k];    // v_cndmask
            const int   ihi = sw ? tix[k] : di;    // v_cndmask
            tv[k] = lo; tix[k] = ilo;
            dv = hi; di = ihi;
          }
        }
      }
      __syncthreads();
    }
    #pragma unroll
    for (int k = 0; k < 9; ++k) cidx[lvl * 9 + k] = tix[k];
  }

  if (half == 0) {
    const float area_g = (gx2 - gx1) * (gy2 - gy1);
    float ciou[27]; int cin[27];
    float s1 = 0.f;
    #pragma unroll
    for (int k = 0; k < 27; ++k) {
      const int an = cidx[k];
      const float* ab = anchors + (size_t)an * 4;
      const float ax1 = ab[0], ay1 = ab[1], ax2 = ab[2], ay2 = ab[3];
      const float iw = fminf(gx2, ax2) - fmaxf(gx1, ax1);
      const float ih = fminf(gy2, ay2) - fmaxf(gy1, ay1);
      const float inter = fmaxf(iw, 0.f) * fmaxf(ih, 0.f);
      const float area_a = (ax2 - ax1) * (ay2 - ay1);
      const float iou = inter / (area_g + area_a - inter + ATSS_EPS);
      ciou[k] = iou; s1 += iou;
      const float acx = 0.5f * (ax1 + ax2), acy = 0.5f * (ay1 + ay2);
      const float mm = fminf(fminf(acx - gx1, acy - gy1),
                             fminf(gx2 - acx, gy2 - acy));
      cin[k] = (mm > ATSS_EPS) ? 1 : 0;
    }
    const float mean = s1 * (1.0f / 27.0f);
    float s2 = 0.f;
    #pragma unroll
    for (int k = 0; k < 27; ++k) { const float dl = ciou[k] - mean; s2 += dl * dl; }
    const float thr = mean + sqrtf(s2 * (1.0f / 26.0f));  // ddof=1

    if (padm[b * n + gi] > 0.5f) {
      #pragma unroll
      for (int k = 0; k < 27; ++k) {
        if (cin[k] && (ciou[k] > thr)) {
          const int an = cidx[k];
          atomicAdd(&cnt[(size_t)b * L + an], 1);
          atomicMin(&firstgt[(size_t)b * L + an], gi);
        }
      }
    }
  }
}

// ---------------------------------------------------------------------------
// Kernel 2: per-anchor resolution + output gather/write.
// ---------------------------------------------------------------------------
__global__ __launch_bounds__(256) void atss_assign_kernel(
    const float* __restrict__ anchors,
    const float* __restrict__ gtb,
    const int*   __restrict__ glab,
    const float* __restrict__ gsc,
    const int*   __restrict__ bgp,
    const int*   __restrict__ cnt,
    const int*   __restrict__ firstgt,
    float* __restrict__ out_lab,
    float* __restrict__ out_box,
    float* __restrict__ out_sco,
    int B, int n, int L, int NC)
{
  const int b = blockIdx.y;
  const int a = blockIdx.x * blockDim.x + threadIdx.x;

  __shared__ float sbox[256 * 4];
  __shared__ int   slab[256];
  __shared__ float sscr[256];
  const int ncache = (n <= 256) ? n : 256;
  for (int i = threadIdx.x; i < ncache; i += blockDim.x) {
    const float* gp = gtb + ((size_t)(b * n + i)) * 4;
    sbox[i * 4 + 0] = gp[0]; sbox[i * 4 + 1] = gp[1];
    sbox[i * 4 + 2] = gp[2]; sbox[i * 4 + 3] = gp[3];
    slab[i] = glab[b * n + i];
    sscr[i] = gsc[b * n + i];
  }
  __syncthreads();
  if (a >= L) return;

  const float* ab = anchors + (size_t)a * 4;
  const float ax1 = ab[0], ay1 = ab[1], ax2 = ab[2], ay2 = ab[3];
  const float area_a = (ax2 - ax1) * (ay2 - ay1);

  const size_t idx = (size_t)b * L + a;
  const int c = cnt[idx];
  const bool pos = (c > 0);
  int gsel = 0;
  if (c > 1) {
    // replaced by is_max_iou: argmax over gts of raw IoU (first-max, like jnp)
    float best = -1.0f; int bi = 0;
    for (int i = 0; i < n; ++i) {
      float x1, y1, x2, y2;
      if (i < ncache) {
        x1 = sbox[i * 4 + 0]; y1 = sbox[i * 4 + 1];
        x2 = sbox[i * 4 + 2]; y2 = sbox[i * 4 + 3];
      } else {
        const float* gp = gtb + ((size_t)(b * n + i)) * 4;
        x1 = gp[0]; y1 = gp[1]; x2 = gp[2]; y2 = gp[3];
      }
      const float iw = fminf(x2, ax2) - fmaxf(x1, ax1);
      const float ih = fminf(y2, ay2) - fmaxf(y1, ay1);
      const float inter = fmaxf(iw, 0.f) * fmaxf(ih, 0.f);
      const float ag = (x2 - x1) * (y2 - y1);
      const float iou = inter / (ag + area_a - inter + ATSS_EPS);
      if (iou > best) { best = iou; bi = i; }
    }
    gsel = bi;
  } else if (c == 1) {
    gsel = firstgt[idx];  // argmax of 0/1 mask = first claiming gt
  } else {
    gsel = 0;             // argmax of all-zero mask = 0 (bbox still gathered!)
  }

  float bx1, by1, bx2, by2; int lab; float sc;
  if (gsel < ncache) {
    bx1 = sbox[gsel * 4 + 0]; by1 = sbox[gsel * 4 + 1];
    bx2 = sbox[gsel * 4 + 2]; by2 = sbox[gsel * 4 + 3];
    lab = slab[gsel]; sc = sscr[gsel];
  } else {
    const float* gp = gtb + ((size_t)(b * n + gsel)) * 4;
    bx1 = gp[0]; by1 = gp[1]; bx2 = gp[2]; by2 = gp[3];
    lab = glab[b * n + gsel]; sc = gsc[b * n + gsel];
  }

  const int bg = bgp[0];
  out_lab[idx] = (float)(pos ? lab : bg);
  out_box[idx * 4 + 0] = bx1;
  out_box[idx * 4 + 1] = by1;
  out_box[idx * 4 + 2] = bx2;
  out_box[idx * 4 + 3] = by2;

  // kept-class column for label: classes != bg kept in order
  int col = -1;
  if (pos) { if (lab < bg) col = lab; else if (lab > bg) col = lab - 1; }
  const float gsv = pos ? sc : 0.0f;

  float* srow = out_sco + idx * (size_t)NC;
  const int nq = NC >> 2;
  float4* srow4 = reinterpret_cast<float4*>(srow);
  for (int q = 0; q < nq; ++q) srow4[q] = make_float4(0.f, 0.f, 0.f, 0.f);
  for (int r = nq << 2; r < NC; ++r) srow[r] = 0.f;
  if (col >= 0) srow[col] = gsv;
}

// ---------------------------------------------------------------------------
extern "C" void kernel_launch(void* const* d_in, const int* in_sizes, int n_in,
                              void* d_out, int out_size, void* d_ws, size_t ws_size,
                              hipStream_t stream) {
  const float* anchors = (const float*)d_in[0];  // [L,4]
  const int*   glab    = (const int*)  d_in[1];  // [B,n,1] int32
  const float* gtb     = (const float*)d_in[2];  // [B,n,4]
  const float* padm    = (const float*)d_in[3];  // [B,n,1]
  const float* gsc     = (const float*)d_in[4];  // [B,n,1]
  const int*   bgp     = (const int*)  d_in[5];  // scalar

  const int L  = in_sizes[0] / 4;                // 8400
  const int BN = in_sizes[1];                    // B*n
  const int NC = 80;                             // NUM_CLASSES (reference const)
  int B = out_size / (L * (5 + NC));             // labels(1)+box(4)+scores(NC)
  if (B <= 0) B = 1;
  const int n = BN / B;

  float* out_lab = (float*)d_out;
  float* out_box = out_lab + (size_t)B * L;
  float* out_sco = out_box + (size_t)B * L * 4;

  int* cnt = (int*)d_ws;
  int* fgt = cnt + (size_t)B * L;
  const int total = B * L;

  atss_init_kernel<<<(total + 255) / 256, 256, 0, stream>>>(cnt, fgt, total);
  atss_topk_kernel<<<B * (n / 16), 32, 0, stream>>>(anchors, gtb, padm,
                                                    cnt, fgt, B, n, L);
  dim3 gridC((L + 255) / 256, B);
  atss_assign_kernel<<<gridC, 256, 0, stream>>>(anchors, gtb, glab, gsc, bgp,
                                                cnt, fgt, out_lab, out_box,
                                                out_sco, B, n, L, NC);
}